// DCN_35373350649921
// MI455X (gfx1250) — compile-verified
//
#include <hip/hip_runtime.h>
#include <math.h>

// Problem constants (from reference setup_inputs)
#define BB   4
#define CC   64
#define HH   128
#define WW   128
#define OO   64
#define KK   9        // 3x3 taps
#define RR   576      // C*K reduction length
#define OMC  27       // offset/mask channels

// workspace layout (float offsets)
#define OM_OFF    0                         // B*27*H*W        = 1,769,472 floats
#define WT2_OFF   (BB * OMC * HH * WW)      // [288][64][2]    =    36,864 floats
#define WOM2_OFF  (WT2_OFF + (RR/2)*OO*2)   // [288][32][2]    =    18,432 floats

typedef __attribute__((ext_vector_type(2))) float v2f;
typedef __attribute__((ext_vector_type(8))) float v8f;

// ---------------------------------------------------------------------------
// Kernel 0: one-time weight repack into WMMA-A-friendly interleaved panels.
//  Wt2 [rp][o][i]  = weight[o*576 + 2*rp + i]           (o in [0,64))
//  Wom2[rp][o][i]  = (o<27) ? om_weight[o*576+2*rp+i]:0 (o in [0,32), zero-pad)
// A fragment then = one coalesced global_load_b64 per lane.
// ---------------------------------------------------------------------------
__global__ __launch_bounds__(256) void dcn_repack(
    const float* __restrict__ weight, const float* __restrict__ omw,
    float* __restrict__ ws) {
  const int e = blockIdx.x * 256 + threadIdx.x;
  const int N1 = (RR / 2) * OO * 2;           // 36864
  const int N2 = (RR / 2) * 32 * 2;           // 18432
  if (e < N1) {
    const int rp  = e >> 7;                   // /128
    const int rem = e & 127;
    const int o   = rem >> 1;
    const int i   = rem & 1;
    ws[WT2_OFF + e] = weight[o * RR + 2 * rp + i];
  } else if (e < N1 + N2) {
    const int e2  = e - N1;
    const int rp  = e2 >> 6;                  // /64
    const int rem = e2 & 63;
    const int o   = rem >> 1;
    const int i   = rem & 1;
    ws[WOM2_OFF + e2] = (o < OMC) ? omw[o * RR + 2 * rp + i] : 0.0f;
  }
}

// ---------------------------------------------------------------------------
// Kernel 1: offset/mask conv as WMMA GEMM (M=27 padded to 32, N=128, K=576).
// Block = one (b,h) row. 8 waves; wave w owns N-tile [16w,16w+16), 2 M-tiles.
// B-panel rows r=c*9+k are plain zero-padded shifted-window reads of x.
// ---------------------------------------------------------------------------
__global__ __launch_bounds__(256) void dcn_om_wmma(
    const float* __restrict__ x, const float* __restrict__ ws,
    const float* __restrict__ omb, float* __restrict__ om) {
  const int bh  = blockIdx.x;
  const int b   = bh >> 7;
  const int h   = bh & 127;
  const int tid = threadIdx.x;
  const float* Wom2 = ws + WOM2_OFF;

  __shared__ float Bpan[2][36 * WW];

  auto produce = [&](int s, int buf) {
    const int c0 = s << 2;
#pragma unroll 2
    for (int it = 0; it < 18; ++it) {
      const int e2 = tid + (it << 8);     // 0..4607
      const int rl = e2 >> 7;             // 0..35
      const int p  = e2 & 127;
      const int cq = rl / 9;
      const int k  = rl - cq * 9;
      const int c  = c0 + cq;
      const int ky = k / 3, kx = k - 3 * (k / 3);
      const int yy = h - 1 + ky;
      const int xx = p - 1 + kx;
      float v = 0.0f;
      if (yy >= 0 && yy < HH && xx >= 0 && xx < WW)
        v = x[(((size_t)b * CC + c) << 14) + (yy << 7) + xx];
      Bpan[buf][(rl << 7) + p] = v;
    }
  };

  const int ln   = tid & 31;
  const int wv   = tid >> 5;
  const int half = ln >> 4;
  const int mrow = ln & 15;
  const int pcol = (wv << 4) + mrow;

  v8f acc[2];
#pragma unroll
  for (int m = 0; m < 2; ++m) acc[m] = (v8f)0.0f;

  auto consume = [&](int s, int buf) {
    const int r0base = s * 36;
    const float* Bp = Bpan[buf];
#pragma unroll
    for (int kb = 0; kb < 9; ++kb) {
      const int rB = (kb << 2) + (half << 1);
      v2f bfrag;
      bfrag.x = Bp[(rB << 7) + pcol];
      bfrag.y = Bp[((rB + 1) << 7) + pcol];
      const int rg = r0base + (kb << 2) + (half << 1);   // even
#pragma unroll
      for (int m = 0; m < 2; ++m) {
        const int o = (m << 4) + mrow;
        const v2f afrag = *(const v2f*)(Wom2 + (size_t)(rg >> 1) * 64 + (o << 1));
        acc[m] = __builtin_amdgcn_wmma_f32_16x16x4_f32(
            false, afrag, false, bfrag, (short)0, acc[m], false, false);
      }
    }
  };

  produce(0, 0);
  __syncthreads();
  for (int s = 0; s < 16; ++s) {
    if (s + 1 < 16) produce(s + 1, (s + 1) & 1);
    consume(s, s & 1);
    __syncthreads();
  }

#pragma unroll
  for (int m = 0; m < 2; ++m) {
#pragma unroll
    for (int j = 0; j < 8; ++j) {
      const int ch = (m << 4) + (half << 3) + j;
      if (ch < OMC)
        om[((((size_t)b * OMC) + ch) << 14) + (h << 7) + pcol] =
            acc[m][j] + omb[ch];
    }
  }
}

// ---------------------------------------------------------------------------
// Kernel 2: fused bilinear sampling + WMMA GEMM (M=64, N=128, K=576).
// Block = one (b,h) row. Double-buffered 36x128 LDS B-panel; per-tap bilinear
// coefficients (channel-invariant) precomputed once in LDS.
// ---------------------------------------------------------------------------
__global__ __launch_bounds__(256) void dcn_main(
    const float* __restrict__ x, const float* __restrict__ ws,
    const float* __restrict__ bias, const float* __restrict__ om,
    float* __restrict__ out) {
  const int bh  = blockIdx.x;
  const int b   = bh >> 7;
  const int h   = bh & 127;
  const int tid = threadIdx.x;
  const float* Wt2 = ws + WT2_OFF;

  __shared__ float tapW[KK * WW][4];   // mask-folded, validity-zeroed corner wts
  __shared__ int   tapYX[KK * WW];     // (y0<<16) | (x0 & 0xffff)
  __shared__ float Bpan[2][36 * WW];   // double-buffered B panel

  // ---- per-tap precompute (channel-invariant) ----
  for (int e = tid; e < KK * WW; e += 256) {
    const int k = e >> 7;
    const int p = e & 127;
    const int ky = k / 3, kx = k - 3 * (k / 3);
    const size_t base = (((size_t)b * OMC) << 14) + (h << 7) + p;
    const float offy = om[base + ((size_t)k << 14)];
    const float offx = om[base + ((size_t)(9 + k) << 14)];
    const float mr   = om[base + ((size_t)(18 + k) << 14)];
    const float mask = 1.0f / (1.0f + __expf(-mr));
    const float py = offy + (float)(h - 1 + ky);
    const float px = offx + (float)(p - 1 + kx);
    const float fy = floorf(py), fx = floorf(px);
    const int   y0 = (int)fy,    x0 = (int)fx;
    const float ly = py - fy, lx = px - fx;
    const float hy = 1.0f - ly, hx = 1.0f - lx;
    const bool vy0 = (y0 >= 0) && (y0 < HH);
    const bool vy1 = (y0 + 1 >= 0) && (y0 + 1 < HH);
    const bool vx0 = (x0 >= 0) && (x0 < WW);
    const bool vx1 = (x0 + 1 >= 0) && (x0 + 1 < WW);
    tapW[e][0] = (vy0 && vx0) ? hy * hx * mask : 0.0f;
    tapW[e][1] = (vy0 && vx1) ? hy * lx * mask : 0.0f;
    tapW[e][2] = (vy1 && vx0) ? ly * hx * mask : 0.0f;
    tapW[e][3] = (vy1 && vx1) ? ly * lx * mask : 0.0f;
    tapYX[e] = (y0 << 16) | (x0 & 0xffff);
  }
  __syncthreads();

  // ---- producer: gather + blend one 36x128 slab (channels 4s..4s+3) ----
  auto produce = [&](int s, int buf) {
    const int c0 = s << 2;
#pragma unroll 2
    for (int it = 0; it < 18; ++it) {
      const int e2 = tid + (it << 8);
      const int rl = e2 >> 7;
      const int p  = e2 & 127;
      const int cq = rl / 9;
      const int k  = rl - cq * 9;
      const int c  = c0 + cq;
      const int tap = (k << 7) | p;
      const int yx  = tapYX[tap];
      const int y0  = yx >> 16;
      const int x0  = (int)(short)(yx & 0xffff);
      const int y0c = min(max(y0, 0), HH - 1);
      const int y1c = min(max(y0 + 1, 0), HH - 1);
      const int x0c = min(max(x0, 0), WW - 1);
      const int x1c = min(max(x0 + 1, 0), WW - 1);
      const float* xb = x + (((size_t)b * CC + c) << 14);
      const float v00 = xb[(y0c << 7) + x0c];
      const float v01 = xb[(y0c << 7) + x1c];
      const float v10 = xb[(y1c << 7) + x0c];
      const float v11 = xb[(y1c << 7) + x1c];
      const float w0 = tapW[tap][0], w1 = tapW[tap][1];
      const float w2 = tapW[tap][2], w3 = tapW[tap][3];
      Bpan[buf][(rl << 7) + p] =
          fmaf(v00, w0, fmaf(v01, w1, fmaf(v10, w2, v11 * w3)));
    }
  };

  // ---- consumer: WMMA over one slab (9 K-blocks of 4, 4 M-tiles) ----
  const int ln   = tid & 31;
  const int wv   = tid >> 5;
  const int half = ln >> 4;
  const int mrow = ln & 15;
  const int pcol = (wv << 4) + mrow;

  v8f acc[4];
#pragma unroll
  for (int m = 0; m < 4; ++m) acc[m] = (v8f)0.0f;

  auto consume = [&](int s, int buf) {
    const int r0base = s * 36;
    const float* Bp = Bpan[buf];
#pragma unroll
    for (int kb = 0; kb < 9; ++kb) {
      const int rB = (kb << 2) + (half << 1);
      v2f bfrag;
      bfrag.x = Bp[(rB << 7) + pcol];         // K = rB,   N = pcol
      bfrag.y = Bp[((rB + 1) << 7) + pcol];   // K = rB+1, N = pcol
      const int rg = r0base + (kb << 2) + (half << 1);   // always even
#pragma unroll
      for (int m = 0; m < 4; ++m) {
        const int o = (m << 4) + mrow;        // A row (output channel)
        const v2f afrag = *(const v2f*)(Wt2 + (size_t)(rg >> 1) * 128 + (o << 1));
        acc[m] = __builtin_amdgcn_wmma_f32_16x16x4_f32(
            false, afrag, false, bfrag, (short)0, acc[m], false, false);
      }
    }
  };

  // ---- double-buffered pipeline over 16 channel-slabs ----
  produce(0, 0);
  __syncthreads();
  for (int s = 0; s < 16; ++s) {
    if (s + 1 < 16) produce(s + 1, (s + 1) & 1);
    consume(s, s & 1);
    __syncthreads();
  }

  // ---- writeback: 16x16 f32 D layout (VGPR j: M = 8*half + j, N = lane&15)
#pragma unroll
  for (int m = 0; m < 4; ++m) {
#pragma unroll
    for (int j = 0; j < 8; ++j) {
      const int o = (m << 4) + (half << 3) + j;
      out[((((size_t)b << 6) + o) << 14) + (h << 7) + pcol] = acc[m][j] + bias[o];
    }
  }
}

// ---------------------------------------------------------------------------
extern "C" void kernel_launch(void* const* d_in, const int* in_sizes, int n_in,
                              void* d_out, int out_size, void* d_ws, size_t ws_size,
                              hipStream_t stream) {
  const float* x      = (const float*)d_in[0];
  const float* weight = (const float*)d_in[1];
  const float* bias   = (const float*)d_in[2];
  const float* om_w   = (const float*)d_in[3];
  const float* om_b   = (const float*)d_in[4];
  float* out = (float*)d_out;
  float* ws  = (float*)d_ws;          // ~7.3 MB: om map + repacked weights
  float* om  = ws + OM_OFF;

  dcn_repack <<<216, 256, 0, stream>>>(weight, om_w, ws);
  dcn_om_wmma<<<BB * HH, 256, 0, stream>>>(x, ws, om_b, om);
  dcn_main   <<<BB * HH, 256, 0, stream>>>(x, ws, bias, om, out);
}